// Model_59528246723048
// MI455X (gfx1250) — compile-verified
//
#include <hip/hip_runtime.h>
#include <math.h>

typedef float v2f __attribute__((ext_vector_type(2)));
typedef float v8f __attribute__((ext_vector_type(8)));

#define ROWS 16384   // B2(8) * NPTS(2048)
#define NPTS 2048
#define MPTS 1024

// ---------------- block reductions (blockDim == 256) ----------------
__device__ __forceinline__ float blkSum(float v, float* red) {
  int t = threadIdx.x;
  red[t] = v; __syncthreads();
  for (int s = 128; s > 0; s >>= 1) { if (t < s) red[t] += red[t + s]; __syncthreads(); }
  float r = red[0]; __syncthreads(); return r;
}
__device__ __forceinline__ float blkMax(float v, float* red) {
  int t = threadIdx.x;
  red[t] = v; __syncthreads();
  for (int s = 128; s > 0; s >>= 1) { if (t < s) red[t] = fmaxf(red[t], red[t + s]); __syncthreads(); }
  float r = red[0]; __syncthreads(); return r;
}
__device__ __forceinline__ float blkMin(float v, float* red) {
  int t = threadIdx.x;
  red[t] = v; __syncthreads();
  for (int s = 128; s > 0; s >>= 1) { if (t < s) red[t] = fminf(red[t], red[t + s]); __syncthreads(); }
  float r = red[0]; __syncthreads(); return r;
}

// ---------------- epilogue helper ----------------
__device__ __forceinline__ float epilogue(float v, float alpha, const float* bias,
                                          const float* resid, int gn,
                                          size_t cidx, int epi) {
  v *= alpha;
  if (bias) v += bias[gn];
  if (epi == 1)      v = fmaxf(v, 0.f);
  else if (epi == 2) v = 0.5f * v * (1.f + erff(v * 0.70710678118654752440f));
  if (resid) v += resid[cidx];
  return v;
}

// ---------------- async global->LDS B128 (gfx1250 ASYNCcnt path) ----------
__device__ __forceinline__ void async_ld128(unsigned lds_off, const float* gptr) {
  unsigned long long ga = (unsigned long long)gptr;
  asm volatile("global_load_async_to_lds_b128 %0, %1, off"
               :: "v"(lds_off), "v"(ga) : "memory");
}

// ---------------- WMMA fp32 GEMM (generic): C = epi(a*A·op(B)+bias)+resid --
// A: MxK row-major (lda). BT=true: B is NxK row-major (weights W[o,c]).
// BT=false: B is KxN row-major (P·V case). epi: 0 none, 1 relu, 2 exact gelu.
template<bool BT>
__global__ __launch_bounds__(256)
void gemm_wmma(const float* __restrict__ A, const float* __restrict__ B,
               float* __restrict__ C, const float* __restrict__ bias,
               const float* __restrict__ resid,
               int M, int N, int K, int lda, int ldb, int ldc,
               float alpha, int epi)
{
  __shared__ float As[64][36];
  __shared__ float Bs[64][36];
  const int tid  = threadIdx.x;
  const int lane = tid & 31;
  const int wave = tid >> 5;   // 0..7
  const int wm   = wave >> 1;  // 0..3  (16-row sub-tile)
  const int wn   = wave & 1;   // 0..1  (32-col sub-tile)
  const int r    = lane & 15;
  const int hi   = lane >> 4;
  const int bm0  = blockIdx.y * 64;
  const int bn0  = blockIdx.x * 64;

  v8f acc0 = {0.f,0.f,0.f,0.f,0.f,0.f,0.f,0.f};
  v8f acc1 = {0.f,0.f,0.f,0.f,0.f,0.f,0.f,0.f};

  for (int k0 = 0; k0 < K; k0 += 32) {
#pragma unroll
    for (int i = 0; i < 8; ++i) {           // 64x32 A tile, 8 elems/thread
      int li = tid + i * 256;
      int m = li >> 5, k = li & 31;
      int gm = bm0 + m, gk = k0 + k;
      As[m][k] = (gm < M && gk < K) ? A[(size_t)gm * lda + gk] : 0.f;
    }
#pragma unroll
    for (int i = 0; i < 8; ++i) {           // 64x32 B tile in [n][k] layout
      int li = tid + i * 256;
      int n = li >> 5, k = li & 31;
      int gn = bn0 + n, gk = k0 + k;
      float v = 0.f;
      if (gn < N && gk < K)
        v = BT ? B[(size_t)gn * ldb + gk] : B[(size_t)gk * ldb + gn];
      Bs[n][k] = v;
    }
    __syncthreads();
#pragma unroll
    for (int kk = 0; kk < 32; kk += 4) {
      v2f a, b0, b1;
      a[0]  = As[wm * 16 + r][kk + 2 * hi + 0];
      a[1]  = As[wm * 16 + r][kk + 2 * hi + 1];
      b0[0] = Bs[wn * 32 + r][kk + 2 * hi + 0];
      b0[1] = Bs[wn * 32 + r][kk + 2 * hi + 1];
      b1[0] = Bs[wn * 32 + 16 + r][kk + 2 * hi + 0];
      b1[1] = Bs[wn * 32 + 16 + r][kk + 2 * hi + 1];
      acc0 = __builtin_amdgcn_wmma_f32_16x16x4_f32(false, a, false, b0, (short)0, acc0, false, false);
      acc1 = __builtin_amdgcn_wmma_f32_16x16x4_f32(false, a, false, b1, (short)0, acc1, false, false);
    }
    __syncthreads();
  }

#pragma unroll
  for (int rr = 0; rr < 8; ++rr) {
    int gm = bm0 + wm * 16 + rr + 8 * hi;
    if (gm >= M) continue;
#pragma unroll
    for (int t = 0; t < 2; ++t) {
      int gn = bn0 + wn * 32 + t * 16 + r;
      if (gn >= N) continue;
      size_t cidx = (size_t)gm * ldc + gn;
      C[cidx] = epilogue(t ? acc1[rr] : acc0[rr], alpha, bias, resid, gn, cidx, epi);
    }
  }
}

// ---------------- WMMA fp32 GEMM, async double-buffered (BT only) ---------
// Requires: M%64==0, N%64==0, K%32==0, lda%4==0, ldb%4==0.
// Global->LDS staging uses GLOBAL_LOAD_ASYNC_TO_LDS_B128 (ASYNCcnt), issuing
// tile k+1 while tile k computes; no VGPR round-trip for operands.
__global__ __launch_bounds__(256)
void gemm_wmma_async(const float* __restrict__ A, const float* __restrict__ B,
                     float* __restrict__ C, const float* __restrict__ bias,
                     const float* __restrict__ resid,
                     int M, int N, int K, int lda, int ldb, int ldc,
                     float alpha, int epi)
{
  __shared__ float As[2][64][36];
  __shared__ float Bs[2][64][36];
  const int tid  = threadIdx.x;
  const int lane = tid & 31;
  const int wave = tid >> 5;
  const int wm   = wave >> 1;
  const int wn   = wave & 1;
  const int r    = lane & 15;
  const int hi   = lane >> 4;
  const int bm0  = blockIdx.y * 64;
  const int bn0  = blockIdx.x * 64;

  // each thread owns 2 A-chunks and 2 B-chunks of 4 floats per tile
  // chunk c (0..511): row m = c>>3, k-offset kc = (c&7)*4  (16B aligned)
  auto issue = [&](int buf, int k0) {
#pragma unroll
    for (int i = 0; i < 2; ++i) {
      int c  = tid + i * 256;
      int m  = c >> 3;
      int kc = (c & 7) * 4;
      async_ld128((unsigned)(size_t)&As[buf][m][kc],
                  A + (size_t)(bm0 + m) * lda + k0 + kc);
      async_ld128((unsigned)(size_t)&Bs[buf][m][kc],
                  B + (size_t)(bn0 + m) * ldb + k0 + kc);
    }
  };

  v8f acc0 = {0.f,0.f,0.f,0.f,0.f,0.f,0.f,0.f};
  v8f acc1 = {0.f,0.f,0.f,0.f,0.f,0.f,0.f,0.f};

  const int nt = K >> 5;
  issue(0, 0);
  for (int it = 0; it < nt; ++it) {
    const int buf = it & 1;
    if (it + 1 < nt) {
      issue(buf ^ 1, (it + 1) << 5);
      asm volatile("s_wait_asynccnt 0x4" ::: "memory");  // this tile's 4 done
    } else {
      asm volatile("s_wait_asynccnt 0x0" ::: "memory");
    }
    __syncthreads();                                      // all waves' data in
#pragma unroll
    for (int kk = 0; kk < 32; kk += 4) {
      v2f a, b0, b1;
      a[0]  = As[buf][wm * 16 + r][kk + 2 * hi + 0];
      a[1]  = As[buf][wm * 16 + r][kk + 2 * hi + 1];
      b0[0] = Bs[buf][wn * 32 + r][kk + 2 * hi + 0];
      b0[1] = Bs[buf][wn * 32 + r][kk + 2 * hi + 1];
      b1[0] = Bs[buf][wn * 32 + 16 + r][kk + 2 * hi + 0];
      b1[1] = Bs[buf][wn * 32 + 16 + r][kk + 2 * hi + 1];
      acc0 = __builtin_amdgcn_wmma_f32_16x16x4_f32(false, a, false, b0, (short)0, acc0, false, false);
      acc1 = __builtin_amdgcn_wmma_f32_16x16x4_f32(false, a, false, b1, (short)0, acc1, false, false);
    }
    __syncthreads();                                      // buf free for refill
  }

#pragma unroll
  for (int rr = 0; rr < 8; ++rr) {
    int gm = bm0 + wm * 16 + rr + 8 * hi;
#pragma unroll
    for (int t = 0; t < 2; ++t) {
      int gn = bn0 + wn * 32 + t * 16 + r;
      size_t cidx = (size_t)gm * ldc + gn;
      C[cidx] = epilogue(t ? acc1[rr] : acc0[rr], alpha, bias, resid, gn, cidx, epi);
    }
  }
}

// ---------------- LayerNorm over rows of 1024 (optional add) ---------------
__global__ __launch_bounds__(256)
void ln_rows(const float* __restrict__ A, const float* __restrict__ Badd,
             float* __restrict__ O, const float* __restrict__ g,
             const float* __restrict__ bta)
{
  __shared__ float red[256];
  const int t = threadIdx.x;
  const size_t base = (size_t)blockIdx.x * 1024;
  float v[4];
  float s = 0.f;
#pragma unroll
  for (int j = 0; j < 4; ++j) {
    int i = t + j * 256;
    float x = A[base + i];
    if (Badd) x += Badd[base + i];
    v[j] = x; s += x;
  }
  float mean = blkSum(s, red) * (1.f / 1024.f);
  float vs = 0.f;
#pragma unroll
  for (int j = 0; j < 4; ++j) { float d = v[j] - mean; vs += d * d; }
  float var  = blkSum(vs, red) * (1.f / 1024.f);
  float rstd = rsqrtf(var + 1e-5f);
#pragma unroll
  for (int j = 0; j < 4; ++j) {
    int i = t + j * 256;
    O[base + i] = (v[j] - mean) * rstd * g[i] + bta[i];
  }
}

// ---------------- in-place row softmax ----------------
__global__ __launch_bounds__(256)
void softmax_rows(float* __restrict__ S, int L)
{
  __shared__ float red[256];
  const int t = threadIdx.x;
  const size_t base = (size_t)blockIdx.x * L;
  float mx = -3.4e38f;
  for (int i = t; i < L; i += 256) mx = fmaxf(mx, S[base + i]);
  mx = blkMax(mx, red);
  float s = 0.f;
  for (int i = t; i < L; i += 256) { float e = expf(S[base + i] - mx); S[base + i] = e; s += e; }
  s = blkSum(s, red);
  float inv = 1.f / s;
  for (int i = t; i < L; i += 256) S[base + i] *= inv;
}

// ---------------- preprocess: com, scale, normalized duplicated x ----------
__global__ __launch_bounds__(256)
void preprocess(const float* __restrict__ x, const float* __restrict__ gt,
                float* __restrict__ fullx, float* __restrict__ scale,
                float* __restrict__ com)
{
  __shared__ float red[256];
  __shared__ float cm[3];
  __shared__ float scl;
  const int b = blockIdx.x;   // 0..3
  const int t = threadIdx.x;
  const size_t gbase = (size_t)b * NPTS * 3;
  float s0 = 0.f, s1 = 0.f, s2 = 0.f;
  for (int n = t; n < NPTS; n += 256) {
    s0 += gt[gbase + n * 3 + 0];
    s1 += gt[gbase + n * 3 + 1];
    s2 += gt[gbase + n * 3 + 2];
  }
  s0 = blkSum(s0, red); s1 = blkSum(s1, red); s2 = blkSum(s2, red);
  if (t == 0) { cm[0] = s0 / NPTS; cm[1] = s1 / NPTS; cm[2] = s2 / NPTS; }
  __syncthreads();
  float c0 = cm[0], c1 = cm[1], c2 = cm[2];
  float mx = 0.f;
  for (int n = t; n < NPTS; n += 256) {
    float dx = x[gbase + n * 3 + 0] - c0;
    float dy = x[gbase + n * 3 + 1] - c1;
    float dz = x[gbase + n * 3 + 2] - c2;
    mx = fmaxf(mx, sqrtf(dx * dx + dy * dy + dz * dz));
  }
  mx = blkMax(mx, red);
  if (t == 0) {
    scl = mx;
    scale[b] = mx; scale[b + 4] = mx;
    com[b * 3 + 0] = c0; com[b * 3 + 1] = c1; com[b * 3 + 2] = c2;
  }
  __syncthreads();
  float inv = 1.f / scl;
  for (int n = t; n < NPTS; n += 256) {
    float vx = (x[gbase + n * 3 + 0] - c0) * inv;
    float vy = (x[gbase + n * 3 + 1] - c1) * inv;
    float vz = (x[gbase + n * 3 + 2] - c2) * inv;
    fullx[gbase + n * 3 + 0] = vx;
    fullx[gbase + n * 3 + 1] = vy;
    fullx[gbase + n * 3 + 2] = vz;
    fullx[(size_t)4 * NPTS * 3 + gbase + n * 3 + 0] = vx;
    fullx[(size_t)4 * NPTS * 3 + gbase + n * 3 + 1] = vy;
    fullx[(size_t)4 * NPTS * 3 + gbase + n * 3 + 2] = vz;
  }
}

// ---------------- head: column softmax over n, fused prediction -----------
__global__ __launch_bounds__(256)
void pred_softmax(const float* __restrict__ h, const float* __restrict__ fullx,
                  const float* __restrict__ scale, const float* __restrict__ com,
                  float* __restrict__ fullpred, float* __restrict__ outpred)
{
  __shared__ float red[256];
  const int m = blockIdx.x, b = blockIdx.y, t = threadIdx.x;
  const float* col = h + (size_t)b * NPTS * 1024 + m;
  float mx = -3.4e38f;
  for (int n = t; n < NPTS; n += 256) mx = fmaxf(mx, col[(size_t)n * 1024]);
  mx = blkMax(mx, red);
  float se = 0.f, sx = 0.f, sy = 0.f, sz = 0.f;
  for (int n = t; n < NPTS; n += 256) {
    float w = expf(col[(size_t)n * 1024] - mx);
    const float* xp = fullx + ((size_t)b * NPTS + n) * 3;
    se += w; sx += w * xp[0]; sy += w * xp[1]; sz += w * xp[2];
  }
  se = blkSum(se, red); sx = blkSum(sx, red); sy = blkSum(sy, red); sz = blkSum(sz, red);
  if (t == 0) {
    float inv = 1.f / se, sc = scale[b];
    const float* cmv = com + (b & 3) * 3;
    float p0 = sx * inv * sc + cmv[0];
    float p1 = sy * inv * sc + cmv[1];
    float p2 = sz * inv * sc + cmv[2];
    size_t o = ((size_t)b * MPTS + m) * 3;
    fullpred[o] = p0; fullpred[o + 1] = p1; fullpred[o + 2] = p2;
    if (b < 4) { outpred[o] = p0; outpred[o + 1] = p1; outpred[o + 2] = p2; }
  }
}

// ---------------- chamfer mins ----------------
__global__ __launch_bounds__(256)
void cd_min_pred(const float* __restrict__ fp, const float* __restrict__ gt,
                 float* __restrict__ d1)
{
  __shared__ float red[256];
  const int m = blockIdx.x, b = blockIdx.y, t = threadIdx.x;
  const float* p = fp + ((size_t)b * MPTS + m) * 3;
  float px = p[0], py = p[1], pz = p[2];
  float pp = px * px + py * py + pz * pz;
  const float* G = gt + (size_t)(b & 3) * NPTS * 3;
  float mn = 3.4e38f;
  for (int n = t; n < NPTS; n += 256) {
    float gx = G[n * 3], gy = G[n * 3 + 1], gz = G[n * 3 + 2];
    float d = pp + gx * gx + gy * gy + gz * gz - 2.f * (px * gx + py * gy + pz * gz);
    mn = fminf(mn, fmaxf(d, 0.f));
  }
  mn = blkMin(mn, red);
  if (t == 0) d1[(size_t)b * MPTS + m] = mn;
}

__global__ __launch_bounds__(256)
void cd_min_gt(const float* __restrict__ fp, const float* __restrict__ gt,
               float* __restrict__ d2m)
{
  __shared__ float red[256];
  const int n = blockIdx.x, b = blockIdx.y, t = threadIdx.x;
  const float* G = gt + (size_t)(b & 3) * NPTS * 3;
  float gx = G[n * 3], gy = G[n * 3 + 1], gz = G[n * 3 + 2];
  float gg = gx * gx + gy * gy + gz * gz;
  const float* P = fp + (size_t)b * MPTS * 3;
  float mn = 3.4e38f;
  for (int m = t; m < MPTS; m += 256) {
    float px = P[m * 3], py = P[m * 3 + 1], pz = P[m * 3 + 2];
    float d = px * px + py * py + pz * pz + gg - 2.f * (px * gx + py * gy + pz * gz);
    mn = fminf(mn, fmaxf(d, 0.f));
  }
  mn = blkMin(mn, red);
  if (t == 0) d2m[(size_t)b * NPTS + n] = mn;
}

__global__ __launch_bounds__(256)
void cd_reduce(const float* __restrict__ d1, const float* __restrict__ d2m,
               float* __restrict__ out)
{
  __shared__ float red[256];
  const int b = blockIdx.x, t = threadIdx.x;
  float s1 = 0.f, s1q = 0.f;
  for (int m = t; m < MPTS; m += 256) { float v = d1[(size_t)b * MPTS + m]; s1 += v; s1q += sqrtf(v); }
  float s2 = 0.f, s2q = 0.f;
  for (int n = t; n < NPTS; n += 256) { float v = d2m[(size_t)b * NPTS + n]; s2 += v; s2q += sqrtf(v); }
  s1 = blkSum(s1, red); s1q = blkSum(s1q, red);
  s2 = blkSum(s2, red); s2q = blkSum(s2q, red);
  if (t == 0) {
    out[12291 + b] = 0.5f * (s1q / MPTS + s2q / NPTS);  // cd_l1
    out[12299 + b] = s1 / MPTS + s2 / NPTS;             // cd_l2
  }
}

// ---------------- k-NN (10 neighbors, JAX top_k stable ties) --------------
__global__ __launch_bounds__(256)
void knn_kernel(const float* __restrict__ fp, int* __restrict__ nbr)
{
  int tg = blockIdx.x * 256 + threadIdx.x;
  if (tg >= 8 * MPTS) return;
  int b = tg >> 10, m = tg & 1023;
  const float* P = fp + (size_t)b * MPTS * 3;
  float px = P[m * 3], py = P[m * 3 + 1], pz = P[m * 3 + 2];
  float pp = px * px + py * py + pz * pz;
  float bd[11]; int bi[11];
#pragma unroll
  for (int k = 0; k < 11; ++k) { bd[k] = 3.4e38f; bi[k] = 0; }
  for (int n = 0; n < MPTS; ++n) {
    float qx = P[n * 3], qy = P[n * 3 + 1], qz = P[n * 3 + 2];
    float d = pp + qx * qx + qy * qy + qz * qz - 2.f * (px * qx + py * qy + pz * qz);
    if (d < bd[10]) {
      int j = 10;
      while (j > 0 && d < bd[j - 1]) { bd[j] = bd[j - 1]; bi[j] = bi[j - 1]; --j; }
      bd[j] = d; bi[j] = n;
    }
  }
#pragma unroll
  for (int k = 0; k < 10; ++k) nbr[(size_t)tg * 10 + k] = bi[k + 1];
}

// ---------------- mapping-error pair matrix ----------------
__global__ __launch_bounds__(256)
void map_err(const float* __restrict__ fp, const int* __restrict__ nbr,
             float* __restrict__ errs)
{
  __shared__ float red[256];
  const int i = blockIdx.x, j = blockIdx.y, t = threadIdx.x;
  const float* Pi = fp + (size_t)i * MPTS * 3;
  const float* Pj = fp + (size_t)j * MPTS * 3;
  float s = 0.f;
  for (int m = t; m < MPTS; m += 256) {
    float pix = Pi[m * 3], piy = Pi[m * 3 + 1], piz = Pi[m * 3 + 2];
    float pjx = Pj[m * 3], pjy = Pj[m * 3 + 1], pjz = Pj[m * 3 + 2];
    const int* nb = nbr + ((size_t)i * MPTS + m) * 10;
#pragma unroll
    for (int k = 0; k < 10; ++k) {
      int n = nb[k];
      float dx = (Pi[n * 3]     - pix) - (Pj[n * 3]     - pjx);
      float dy = (Pi[n * 3 + 1] - piy) - (Pj[n * 3 + 1] - pjy);
      float dz = (Pi[n * 3 + 2] - piz) - (Pj[n * 3 + 2] - pjz);
      s += dx * dx + dy * dy + dz * dz;
    }
  }
  s = blkSum(s, red);
  if (t == 0) errs[i * 8 + j] = s / 10240.f;
}

// ---------------- final scalars ----------------
__global__ __launch_bounds__(256)
void finalize(const float* __restrict__ fp, const float* __restrict__ errs,
              const int* __restrict__ labels, const int* __restrict__ epoch,
              float* __restrict__ out)
{
  __shared__ float red[256];
  const int t = threadIdx.x;
  float s = 0.f;
  for (int i = t; i < 12288; i += 256) { float d = fp[i] - fp[12288 + i]; s += d * d; }
  s = blkSum(s, red);
  if (t == 0) {
    float consist = s / 12288.f * 1024.f;
    float me_s = 0.f; int cnt = 0;
    for (int i = 0; i < 8; ++i)
      for (int j = 0; j < 8; ++j)
        if (i != j && labels[i & 3] == labels[j & 3]) { ++cnt; me_s += errs[i * 8 + j]; }
    float me = (cnt > 0) ? me_s / (float)cnt : 0.f;
    float cdsum = 0.f;
    for (int b = 0; b < 8; ++b) cdsum += out[12299 + b];
    float base = cdsum / 8.f + 0.1f * me;
    float loss = (epoch[0] < 10) ? base : base + consist;
    out[12288] = loss;
    out[12289] = base;
    out[12290] = consist;
    out[12307] = me;
  }
}

// ---------------- host side ----------------
static void gemm(bool bt, hipStream_t st, const float* A, const float* B, float* C,
                 const float* bias, const float* resid,
                 int M, int N, int K, int lda, int ldb, int ldc, float alpha, int epi)
{
  dim3 g((N + 63) / 64, (M + 63) / 64), blk(256);
  if (bt && (K % 32 == 0) && (M % 64 == 0) && (N % 64 == 0) &&
      (lda % 4 == 0) && (ldb % 4 == 0)) {
    gemm_wmma_async<<<g, blk, 0, st>>>(A, B, C, bias, resid, M, N, K, lda, ldb, ldc, alpha, epi);
  } else if (bt) {
    gemm_wmma<true ><<<g, blk, 0, st>>>(A, B, C, bias, resid, M, N, K, lda, ldb, ldc, alpha, epi);
  } else {
    gemm_wmma<false><<<g, blk, 0, st>>>(A, B, C, bias, resid, M, N, K, lda, ldb, ldc, alpha, epi);
  }
}

extern "C" void kernel_launch(void* const* d_in, const int* in_sizes, int n_in,
                              void* d_out, int out_size, void* d_ws, size_t ws_size,
                              hipStream_t stream)
{
  (void)in_sizes; (void)n_in; (void)out_size; (void)ws_size;
  // input order: x, gt, labels, [params flattened depth-first, sorted keys], epoch
  const float* x      = (const float*)d_in[0];
  const float* gt     = (const float*)d_in[1];
  const int*   labels = (const int*)d_in[2];
  const float* eb1 = (const float*)d_in[3];
  const float* eb2 = (const float*)d_in[4];
  const float* eb3 = (const float*)d_in[5];
  const float* ew1 = (const float*)d_in[6];
  const float* ew2 = (const float*)d_in[7];
  const float* ew3 = (const float*)d_in[8];
  const float* SA[3][14];
  for (int blk = 0; blk < 3; ++blk)
    for (int i = 0; i < 14; ++i)
      SA[blk][i] = (const float*)d_in[9 + 14 * blk + i];
  const int* epoch = (const int*)d_in[51];
  float* out = (float*)d_out;

  // ---- workspace layout (~487 MB of f32) ----
  float* ws = (float*)d_ws;
  float* U0 = ws;                       // 16384x1024
  float* U1 = U0 + (size_t)ROWS * 1024;
  float* U2 = U1 + (size_t)ROWS * 1024;
  float* U3 = U2 + (size_t)ROWS * 1024;
  float* UQ = U3 + (size_t)ROWS * 1024; // 16384x3072
  float* Sb = UQ + (size_t)ROWS * 3072; // 2048x2048 attention scratch
  float* fullx    = Sb + (size_t)NPTS * NPTS;     // 8x2048x3
  float* fullpred = fullx + (size_t)8 * NPTS * 3; // 8x1024x3
  float* scale    = fullpred + (size_t)8 * MPTS * 3; // 8
  float* com      = scale + 8;                       // 12
  float* d1buf    = com + 12;                        // 8x1024
  float* d2mbuf   = d1buf + (size_t)8 * MPTS;        // 8x2048
  float* errs     = d2mbuf + (size_t)8 * NPTS;       // 64
  int*   nbr      = (int*)(errs + 64);               // 8x1024x10

  // ---- preprocess ----
  preprocess<<<4, 256, 0, stream>>>(x, gt, fullx, scale, com);

  // ---- encoder (H1->U0, H2->U1, feats->U2) ----
  gemm(true, stream, fullx, ew1, U0, eb1, nullptr, ROWS,  64,   3,   3,   3,  64, 1.f, 1);
  gemm(true, stream, U0,    ew2, U1, eb2, nullptr, ROWS, 128,  64,  64,  64, 128, 1.f, 1);
  gemm(true, stream, U1,    ew3, U2, eb3, nullptr, ROWS, 128, 128, 128, 128, 128, 1.f, 0);

  // ---- 3 transformer blocks ----
  const float* in = U2; int din = 128;
  for (int blk = 0; blk < 3; ++blk) {
    const float* in_b   = SA[blk][0];  const float* in_w   = SA[blk][1];
    const float* l11_b  = SA[blk][2];  const float* l11_w  = SA[blk][3];
    const float* l12_b  = SA[blk][4];  const float* l12_w  = SA[blk][5];
    const float* n12_b  = SA[blk][6];  const float* n12_g  = SA[blk][7];
    const float* n13_b  = SA[blk][8];  const float* n13_g  = SA[blk][9];
    const float* out_b  = SA[blk][10]; const float* out_w  = SA[blk][11];
    const float* proj_b = SA[blk][12]; const float* proj_w = SA[blk][13];

    // proj + LN13 -> s1n in U0
    gemm(true, stream, in, proj_w, U0, proj_b, nullptr, ROWS, 1024, din, din, din, 1024, 1.f, 0);
    ln_rows<<<ROWS, 256, 0, stream>>>(U0, nullptr, U0, n13_g, n13_b);
    // QKV -> UQ
    gemm(true, stream, U0, in_w, UQ, in_b, nullptr, ROWS, 3072, 1024, 1024, 1024, 3072, 1.f, 0);
    // attention per (b, h): S = QK^T/16, softmax, O = S·V  -> U1
    for (int b = 0; b < 8; ++b) {
      for (int h = 0; h < 4; ++h) {
        const float* Qp = UQ + (size_t)b * NPTS * 3072 + h * 256;
        const float* Kp = Qp + 1024;
        const float* Vp = Qp + 2048;
        gemm(true, stream, Qp, Kp, Sb, nullptr, nullptr,
             NPTS, NPTS, 256, 3072, 3072, NPTS, 0.0625f, 0);
        softmax_rows<<<NPTS, 256, 0, stream>>>(Sb, NPTS);
        float* Op = U1 + (size_t)b * NPTS * 1024 + h * 256;
        gemm(false, stream, Sb, Vp, Op, nullptr, nullptr,
             NPTS, 256, NPTS, NPTS, 3072, 1024, 1.f, 0);
      }
    }
    // out-proj -> U2 ; s1 = LN12(s1n + AO) -> U1
    gemm(true, stream, U1, out_w, U2, out_b, nullptr, ROWS, 1024, 1024, 1024, 1024, 1024, 1.f, 0);
    ln_rows<<<ROWS, 256, 0, stream>>>(U0, U2, U1, n12_g, n12_b);
    // FF: gelu(s1·l11^T) -> UQ ; out = UQ·l12^T + b + s1 -> U3
    gemm(true, stream, U1, l11_w, UQ, l11_b, nullptr, ROWS, 1024, 1024, 1024, 1024, 1024, 1.f, 2);
    gemm(true, stream, UQ, l12_w, U3, l12_b, U1,      ROWS, 1024, 1024, 1024, 1024, 1024, 1.f, 0);
    in = U3; din = 1024;
  }

  // ---- head: softmax over points + prediction (writes pred to d_out) ----
  pred_softmax<<<dim3(MPTS, 8), 256, 0, stream>>>(U3, fullx, scale, com, fullpred, out);

  // ---- chamfer ----
  cd_min_pred<<<dim3(MPTS, 8), 256, 0, stream>>>(fullpred, gt, d1buf);
  cd_min_gt  <<<dim3(NPTS, 8), 256, 0, stream>>>(fullpred, gt, d2mbuf);
  cd_reduce  <<<8, 256, 0, stream>>>(d1buf, d2mbuf, out);

  // ---- mapping error + final scalars ----
  knn_kernel<<<(8 * MPTS + 255) / 256, 256, 0, stream>>>(fullpred, nbr);
  map_err<<<dim3(8, 8), 256, 0, stream>>>(fullpred, nbr, errs);
  finalize<<<1, 256, 0, stream>>>(fullpred, errs, labels, epoch, out);
}